// NLSA_6262062317891
// MI455X (gfx1250) — compile-verified
//
#include <hip/hip_runtime.h>
#include <stdint.h>

typedef __attribute__((ext_vector_type(2))) float        v2f;
typedef __attribute__((ext_vector_type(4))) float        v4f;
typedef __attribute__((ext_vector_type(8))) float        v8f;
typedef __attribute__((ext_vector_type(4))) int          v4i;
typedef __attribute__((ext_vector_type(8))) int          v8i;
typedef __attribute__((ext_vector_type(4))) unsigned int v4u;

#define N_B   16
#define C_DIM 64
#define P_DIM 65536     // H*W = 256*256
#define M_DIM 128
#define LDS_STRIDE 144  // 128 + 16 pad dwords: matches TDM pad_interval=128dw, pad_amount=16dw

#if defined(__has_builtin)
#if __has_builtin(__builtin_amdgcn_tensor_load_to_lds) && __has_builtin(__builtin_amdgcn_s_wait_tensorcnt)
#define HAVE_TDM 1
#endif
#if __has_builtin(__builtin_amdgcn_global_load_async_to_lds_b128)
#define HAVE_ASYNC_LDS 1
#endif
#if __has_builtin(__builtin_amdgcn_s_wait_asynccnt)
#define HAVE_WAIT_ASYNC 1
#endif
#endif
#ifndef HAVE_TDM
#define HAVE_TDM 0
#endif
#ifndef HAVE_ASYNC_LDS
#define HAVE_ASYNC_LDS 0
#endif
#ifndef HAVE_WAIT_ASYNC
#define HAVE_WAIT_ASYNC 0
#endif

typedef __attribute__((address_space(1))) v4i  g_v4i;
typedef __attribute__((address_space(3))) v4i  l_v4i;
typedef __attribute__((address_space(3))) void l_void;

// out[n, p, m] = sum_c x[n, c, p] * rm[n, c, m]
// One wave -> 16 (p) x 128 (m) output strip: 8 WMMA tiles, K-loop of 16 steps (K=4 each).
__global__ __launch_bounds__(256) void lsh_hash_wmma_f32(
    const float* __restrict__ x,    // (N, C, P)
    const float* __restrict__ rm,   // (N, C, M)
    float* __restrict__ out)        // (N, P, M)
{
    __shared__ float sB[C_DIM * LDS_STRIDE];   // 36 KB, rm[n] with padded rows

    const int tid  = threadIdx.x;
    const int lane = tid & 31;
    const int wv   = tid >> 5;
    const int n    = blockIdx.y;

    // ---------- stage rm[n] (64 rows x 128 f32) into LDS, row stride 144 dwords ----------
    const float* rm_n = rm + (size_t)n * C_DIM * M_DIM;
#if HAVE_TDM
    if (wv == 0) {
        const uint64_t ga  = (uint64_t)(uintptr_t)rm_n;             // global byte addr
        const uint32_t la  = (uint32_t)(size_t)(l_void*)sB;         // LDS byte offset
        // ---- D# group 0: count=1 | lds_addr | global_addr[56:0] | type=2 ----
        v4u g0;
        g0.x = 1u;                                                  // count=1, user desc
        g0.y = la;                                                  // lds_addr
        g0.z = (uint32_t)ga;                                        // global_addr[31:0]
        g0.w = (uint32_t)((ga >> 32) & 0x1FFFFFFu) | (2u << 30);    // global_addr[56:32] | type=2
        // ---- D# group 1 ----
        v8i g1;
        g1[0] = (int)((2u << 16)        // data_size = 4 bytes
                    | (1u << 20)        // pad_enable
                    | (6u << 22)        // pad_interval: 128 dwords (one 512B row)
                    | (15u << 25));     // pad_amount: 16 dwords
        g1[1] = (int)((uint32_t)M_DIM << 16);   // tensor_dim0[15:0]=128 (atomic_barrier_addr=0)
        g1[2] = (int)((uint32_t)C_DIM << 16);   // tensor_dim0[31:16]=0, tensor_dim1[15:0]=64
        g1[3] = (int)((uint32_t)M_DIM << 16);   // tensor_dim1[31:16]=0, tile_dim0=128
        g1[4] = (int)C_DIM;                     // tile_dim1=64, tile_dim2=0
        g1[5] = (int)M_DIM;                     // tensor_dim0_stride[31:0]=128
        g1[6] = (int)((uint32_t)(M_DIM * C_DIM) << 16); // dim0_stride[47:32]=0, dim1_stride[15:0]
        g1[7] = 0;                              // tensor_dim1_stride[47:16]=0
        // ---- groups 2/3/4 unused (2D tile: tile_dim2..4 = 0) ----
        v4i g2 = {0, 0, 0, 0};
        v4i g3 = {0, 0, 0, 0};
        v8i g4 = {0, 0, 0, 0, 0, 0, 0, 0};
        __builtin_amdgcn_tensor_load_to_lds(g0, g1, g2, g3, g4, 0);
        __builtin_amdgcn_s_wait_tensorcnt(0);
    }
#elif HAVE_ASYNC_LDS
    #pragma unroll
    for (int i = 0; i < 8; ++i) {
        const int q  = tid + i * 256;   // float4 index 0..2047
        const int c  = q >> 5;          // 32 float4 per row
        const int m4 = q & 31;
        __builtin_amdgcn_global_load_async_to_lds_b128(
            (g_v4i*)(rm_n + c * M_DIM + m4 * 4),
            (l_v4i*)(&sB[c * LDS_STRIDE + m4 * 4]),
            0, 0);
    }
#if HAVE_WAIT_ASYNC
    __builtin_amdgcn_s_wait_asynccnt(0);
#else
    asm volatile("s_wait_asynccnt 0" ::: "memory");
#endif
#else
    #pragma unroll
    for (int i = 0; i < 8; ++i) {
        const int q  = tid + i * 256;
        const int c  = q >> 5;
        const int m4 = q & 31;
        *(v4f*)(&sB[c * LDS_STRIDE + m4 * 4]) =
            *(const v4f*)(rm_n + c * M_DIM + m4 * 4);
    }
#endif
    __syncthreads();

    // ---------- WMMA main loop ----------
    const int p_base = (blockIdx.x * 8 + wv) * 16;
    const int prow   = lane & 15;   // pixel row within tile (A M-index, B/D N-index)
    const int khalf  = lane >> 4;   // 0: K=0,1  1: K=2,3 within each K=4 step

    const float* Xn = x + (size_t)n * C_DIM * P_DIM + p_base + prow;

    v8f acc[8];
    #pragma unroll
    for (int m = 0; m < 8; ++m) acc[m] = (v8f)0.0f;

    #pragma unroll
    for (int t = 0; t < 16; ++t) {
        const int c0 = 4 * t + 2 * khalf;
        // A fragment: 16x4 f32  (lane l<16: K=0,1 ; l>=16: K=2,3)
        v2f a;
        a.x = Xn[(size_t)c0 * P_DIM];
        a.y = Xn[(size_t)(c0 + 1) * P_DIM];
        const float* b0 = &sB[c0 * LDS_STRIDE + prow];
        const float* b1 = &sB[(c0 + 1) * LDS_STRIDE + prow];
        #pragma unroll
        for (int m = 0; m < 8; ++m) {
            // B fragment: 4x16 f32, mirrored layout from LDS
            v2f b;
            b.x = b0[m * 16];
            b.y = b1[m * 16];
            acc[m] = __builtin_amdgcn_wmma_f32_16x16x4_f32(
                false, a, false, b, (short)0, acc[m], false, false);
        }
    }

    // ---------- store D tiles (VGPR r: lanes 0-15 -> M=r, lanes 16-31 -> M=8+r) ----------
    float* On = out + (size_t)n * P_DIM * M_DIM + (size_t)p_base * M_DIM;
    const int mrow = 8 * khalf;
    #pragma unroll
    for (int m = 0; m < 8; ++m) {
        #pragma unroll
        for (int r = 0; r < 8; ++r) {
            On[(size_t)(r + mrow) * M_DIM + m * 16 + prow] = acc[m][r];
        }
    }
}

extern "C" void kernel_launch(void* const* d_in, const int* in_sizes, int n_in,
                              void* d_out, int out_size, void* d_ws, size_t ws_size,
                              hipStream_t stream) {
    const float* x   = (const float*)d_in[0];   // (16, 64, 256, 256) fp32
    const float* rm  = (const float*)d_in[1];   // (16, 64, 128) fp32
    float* out = (float*)d_out;                 // (16, 65536, 128) fp32

    dim3 grid(P_DIM / 128, N_B);                // 512 x 16 blocks
    lsh_hash_wmma_f32<<<grid, 256, 0, stream>>>(x, rm, out);
}